// GCN_1168231104918
// MI455X (gfx1250) — compile-verified
//
#include <hip/hip_runtime.h>
#include <hip/hip_bf16.h>

typedef __attribute__((ext_vector_type(2))) float v2f;
typedef __attribute__((ext_vector_type(8))) float v8f;

// ---------------------------------------------------------------------------
// Utility fill
// ---------------------------------------------------------------------------
__global__ void gcn_fill(float* __restrict__ p, int n, float v) {
    int i = blockIdx.x * blockDim.x + threadIdx.x;
    if (i < n) p[i] = v;
}

// ---------------------------------------------------------------------------
// Degree (atomic count of in-edges; deg pre-initialized to 1.0 for self-loop)
// ---------------------------------------------------------------------------
__global__ void gcn_deg(const int* __restrict__ dst, int E, float* __restrict__ deg) {
    int e = blockIdx.x * blockDim.x + threadIdx.x;
    if (e < E) atomicAdd(&deg[dst[e]], 1.0f);
}

__global__ void gcn_dinv(float* __restrict__ deg, int n) {
    int i = blockIdx.x * blockDim.x + threadIdx.x;
    if (i < n) {
        float d = deg[i];
        deg[i] = (d > 0.0f) ? rsqrtf(d) : 0.0f;
    }
}

// ---------------------------------------------------------------------------
// WMMA fp32 GEMM: C[M x NC] = A[M x 64] @ B[64 x NC], NC compile-time.
// One wave per 16x16 output tile; K stepped by 4 with V_WMMA_F32_16X16X4_F32.
// All loads/stores use a single per-lane base pointer + immediate offsets
// (max B offset 63*NC*4 <= 16128 B, inside the 24-bit IOFFSET field).
// A layout (16x4, 2 VGPRs/lane): lanes 0-15 -> M=lane, K={k,k+1};
//                                lanes 16-31 -> M=lane-16, K={k+2,k+3}.
// B layout (4x16): mirrored; C/D: VGPR r -> M=r (+8 for upper half-wave), N=lane&15.
// ---------------------------------------------------------------------------
template <int NC>
__global__ void gcn_gemm_wmma(const float* __restrict__ A, const float* __restrict__ B,
                              float* __restrict__ C, int rowTiles) {
    constexpr int K = 64;
    constexpr int colTiles = NC >> 4;
    const int wave = threadIdx.x >> 5;
    const int lane = threadIdx.x & 31;
    const int gw = blockIdx.x * (blockDim.x >> 5) + wave;
    const int rowTile = gw / colTiles;
    const int colTile = gw - rowTile * colTiles;
    if (rowTile >= rowTiles) return;  // wave-uniform exit: EXEC stays all-1s

    const int l15 = lane & 15;
    const int kh  = (lane >> 4) << 1;            // 0 for lanes 0-15, 2 for 16-31

    // Per-lane base pointers; every access below is base + constant offset.
    const float* __restrict__ arow = A + (size_t)(rowTile * 16 + l15) * K + kh;
    const float* __restrict__ bcol = B + (size_t)kh * NC + colTile * 16 + l15;
    float* __restrict__       crow =
        C + (size_t)(rowTile * 16 + ((lane >> 4) << 3)) * NC + colTile * 16 + l15;

    v8f c = {};
#pragma unroll
    for (int k0 = 0; k0 < K; k0 += 4) {
        v2f a = *reinterpret_cast<const v2f*>(arow + k0);   // 8B-aligned b64 load
        v2f b;
        b.x = bcol[k0 * NC];                                // imm offset k0*NC*4
        b.y = bcol[(k0 + 1) * NC];                          // imm offset (k0+1)*NC*4
        c = __builtin_amdgcn_wmma_f32_16x16x4_f32(
            /*neg_a=*/false, a, /*neg_b=*/false, b,
            /*c_mod=*/(short)0, c, /*reuse_a=*/false, /*reuse_b=*/false);
    }

#pragma unroll
    for (int r = 0; r < 8; ++r) crow[r * NC] = c[r];        // imm offset r*NC*4
}

// ---------------------------------------------------------------------------
// Edge scatter (layer 1): 64 features, one wave per edge, 2 feats/lane (float2)
// ---------------------------------------------------------------------------
__global__ void gcn_scatter64(const int* __restrict__ src, const int* __restrict__ dst,
                              int E, const float* __restrict__ dinv,
                              const float* __restrict__ h, float* __restrict__ agg) {
    long t = (long)blockIdx.x * blockDim.x + threadIdx.x;
    int e = (int)(t >> 5);
    if (e >= E) return;
    int lane = (int)(t & 31);
    int s = src[e], d = dst[e];
    float w = dinv[s] * dinv[d];
    v2f v = *reinterpret_cast<const v2f*>(h + (size_t)s * 64 + lane * 2);
    float* o = agg + (size_t)d * 64 + lane * 2;
    atomicAdd(o,     w * v.x);
    atomicAdd(o + 1, w * v.y);
}

// ---------------------------------------------------------------------------
// Post layer-1: self-loop term + bias + ReLU (in place on agg1)
// ---------------------------------------------------------------------------
__global__ void gcn_post1(float* __restrict__ agg, const float* __restrict__ h,
                          const float* __restrict__ dinv, const float* __restrict__ b,
                          int total) {
    int i = blockIdx.x * blockDim.x + threadIdx.x;
    if (i >= total) return;
    int node = i >> 6;
    int f = i & 63;
    float di = dinv[node];
    float v = agg[i] + di * di * h[i] + b[f];
    agg[i] = fmaxf(v, 0.0f);
}

// ---------------------------------------------------------------------------
// Edge scatter (layer 2): 16 features, half-wave per edge
// ---------------------------------------------------------------------------
__global__ void gcn_scatter16(const int* __restrict__ src, const int* __restrict__ dst,
                              int E, const float* __restrict__ dinv,
                              const float* __restrict__ h, float* __restrict__ out) {
    long t = (long)blockIdx.x * blockDim.x + threadIdx.x;
    int e = (int)(t >> 4);
    if (e >= E) return;
    int lane = (int)(t & 15);
    int s = src[e], d = dst[e];
    float w = dinv[s] * dinv[d];
    atomicAdd(&out[(size_t)d * 16 + lane], w * h[(size_t)s * 16 + lane]);
}

// ---------------------------------------------------------------------------
// Post layer-2: self-loop + bias, then row-wise log_softmax over 16 cols.
// 16 lanes per node; reductions via shfl_xor within 16-lane groups (wave32).
// ---------------------------------------------------------------------------
__global__ void gcn_lsm(float* __restrict__ out, const float* __restrict__ h,
                        const float* __restrict__ dinv, const float* __restrict__ b,
                        int N) {
    int t = blockIdx.x * blockDim.x + threadIdx.x;
    int node = t >> 4;
    if (node >= N) return;
    int f = t & 15;
    size_t idx = (size_t)node * 16 + f;
    float di = dinv[node];
    float v = out[idx] + di * di * h[idx] + b[f];

    float m = v;
#pragma unroll
    for (int off = 8; off > 0; off >>= 1) m = fmaxf(m, __shfl_xor(m, off, 16));
    float ex = __expf(v - m);
    float s = ex;
#pragma unroll
    for (int off = 8; off > 0; off >>= 1) s += __shfl_xor(s, off, 16);
    out[idx] = (v - m) - __logf(s);
}

// ---------------------------------------------------------------------------
// Launch
// ---------------------------------------------------------------------------
extern "C" void kernel_launch(void* const* d_in, const int* in_sizes, int n_in,
                              void* d_out, int out_size, void* d_ws, size_t ws_size,
                              hipStream_t stream) {
    const float* x  = (const float*)d_in[0];
    const int*   ei = (const int*)d_in[1];   // [2, E]: row0 = src, row1 = dst
    const float* W1 = (const float*)d_in[2];
    const float* b1 = (const float*)d_in[3];
    const float* W2 = (const float*)d_in[4];
    const float* b2 = (const float*)d_in[5];

    const int N = in_sizes[0] / 64;          // 50000
    const int E = in_sizes[1] / 2;           // 800000
    const int* src = ei;
    const int* dst = ei + E;

    float* deg  = (float*)d_ws;              // N (holds dinv after gcn_dinv)
    float* h1   = deg  + 51200;              // N*64
    float* agg1 = h1   + (size_t)N * 64;     // N*64
    float* h2   = agg1 + (size_t)N * 64;     // N*16
    float* out  = (float*)d_out;             // N*16 (accumulator, then final)

    const int B256 = 256;

    // Init (must happen every call; harness does not re-zero buffers)
    gcn_fill<<<(N * 64 + B256 - 1) / B256, B256, 0, stream>>>(agg1, N * 64, 0.0f);
    gcn_fill<<<(N * 16 + B256 - 1) / B256, B256, 0, stream>>>(out, N * 16, 0.0f);
    gcn_fill<<<(N + B256 - 1) / B256, B256, 0, stream>>>(deg, N, 1.0f);  // self-loop

    // Degree -> dinv
    gcn_deg<<<(E + B256 - 1) / B256, B256, 0, stream>>>(dst, E, deg);
    gcn_dinv<<<(N + B256 - 1) / B256, B256, 0, stream>>>(deg, N);

    // GEMM1: h1 = x @ W1   (50000x64 @ 64x64), 4 col tiles per row tile
    const int rowTiles = N / 16;             // 3125 exactly
    {
        int waves = rowTiles * 4;
        int blocks = (waves + 7) / 8;        // 8 waves per 256-thread block
        gcn_gemm_wmma<64><<<blocks, 256, 0, stream>>>(x, W1, h1, rowTiles);
    }

    // Aggregate layer 1
    gcn_scatter64<<<(int)(((long)E * 32 + B256 - 1) / B256), B256, 0, stream>>>(
        src, dst, E, deg, h1, agg1);
    gcn_post1<<<(N * 64 + B256 - 1) / B256, B256, 0, stream>>>(agg1, h1, deg, b1, N * 64);

    // GEMM2: h2 = relu_out @ W2   (50000x64 @ 64x16), 1 col tile per row tile
    {
        int waves = rowTiles;
        int blocks = (waves + 7) / 8;
        gcn_gemm_wmma<16><<<blocks, 256, 0, stream>>>(agg1, W2, h2, rowTiles);
    }

    // Aggregate layer 2 into d_out, then fused self-loop + bias + log_softmax
    gcn_scatter16<<<(int)(((long)E * 16 + B256 - 1) / B256), B256, 0, stream>>>(
        src, dst, E, deg, h2, out);
    gcn_lsm<<<(N * 16 + B256 - 1) / B256, B256, 0, stream>>>(out, h2, deg, b2, N);
}